// T2SStageDecoder_71966472012567
// MI455X (gfx1250) — compile-verified
//
#include <hip/hip_runtime.h>
#include <math.h>
#include <stdint.h>

// ---------------- model constants ----------------
#define LAYERS 24
#define DMODEL 512
#define HEADS  16
#define HDIM   32
#define FFNDIM 2048
#define VOCAB  1025
#define SKV    2048        // padded cache length
#define SIN    2047        // input cache length
#define TY     1535
#define NTOK   1536        // y length (Ty+1)

typedef __attribute__((ext_vector_type(2))) float v2f;
typedef __attribute__((ext_vector_type(8))) float v8f;
typedef int v4i __attribute__((vector_size(16)));   // matches builtin's expected pointee

#define NEG_INF (-__builtin_huge_valf())

// ---- gfx1250 async LDS-DMA helpers (ASYNCcnt path) ----
__device__ __forceinline__ void async_copy_b128(const float* g, float* l) {
#if defined(__has_builtin) && __has_builtin(__builtin_amdgcn_global_load_async_to_lds_b128)
    __builtin_amdgcn_global_load_async_to_lds_b128(
        (__attribute__((address_space(1))) v4i*)(uintptr_t)g,
        (__attribute__((address_space(3))) v4i*)(uint32_t)(uintptr_t)l,
        0, 0);
#else
    *(float4*)l = *(const float4*)g;
#endif
}

__device__ __forceinline__ void async_wait0() {
#if defined(__has_builtin) && __has_builtin(__builtin_amdgcn_s_wait_asynccnt)
    __builtin_amdgcn_s_wait_asynccnt(0);
#else
    asm volatile("s_wait_asynccnt 0x0" ::: "memory");
#endif
}

// ---------------- prep: x = emb[y_last] + alpha*pe[1535]; also last row of y_emb_new ---------
__global__ void prep_x_kernel(const int* __restrict__ y, const float* __restrict__ emb,
                              const float* __restrict__ alpha,
                              float* __restrict__ x, float* __restrict__ yemb_last) {
    int t = threadIdx.x;               // 512 threads
    int tok = y[NTOK - 1];
    float e = emb[(size_t)tok * DMODEL + t];
    int i2 = t & ~1;
    float div = __expf((float)i2 * (-9.210340371976184f / (float)DMODEL)); // -ln(10000)/D
    float ang = (float)TY * div;
    float pe = (t & 1) ? __cosf(ang) : __sinf(ang);
    x[t] = e + alpha[0] * pe;
    yemb_last[t] = e;
}

// ---------------- generic copies ----------------
__global__ void copy_f4_kernel(const float4* __restrict__ src, float4* __restrict__ dst, int n4) {
    for (int i = blockIdx.x * blockDim.x + threadIdx.x; i < n4; i += gridDim.x * blockDim.x)
        dst[i] = src[i];
}

__global__ void copy_i2f_kernel(const int* __restrict__ src, float* __restrict__ dst, int n) {
    for (int i = blockIdx.x * blockDim.x + threadIdx.x; i < n; i += gridDim.x * blockDim.x)
        dst[i] = (float)src[i];
}

// KV cache pad-copy: (L, 2047, 512) -> (L, 2048, 512), float4 granularity
__global__ void copy_kv_kernel(const float4* __restrict__ src, float4* __restrict__ dst) {
    const size_t per_src = (size_t)SIN * (DMODEL / 4);   // 262016
    const size_t per_dst = (size_t)SKV * (DMODEL / 4);   // 262144
    const size_t total = (size_t)LAYERS * per_src;
    for (size_t i = blockIdx.x * (size_t)blockDim.x + threadIdx.x; i < total;
         i += (size_t)gridDim.x * blockDim.x) {
        size_t l = i / per_src;
        size_t r = i - l * per_src;
        dst[l * per_dst + r] = src[i];
    }
}

// scatter new k/v rows into cache slot 2047
__global__ void append_kv_kernel(const float* __restrict__ qkv,
                                 float* __restrict__ kdst, float* __restrict__ vdst) {
    int t = threadIdx.x;               // 512 threads, 2 blocks
    if (blockIdx.x == 0) kdst[t] = qkv[DMODEL + t];
    else                 vdst[t] = qkv[2 * DMODEL + t];
}

// ---------------- WMMA GEMV: y[e] = dot(W[e,:], x) + bias[e], optional relu -------------
// One wave computes a 16-row output tile via V_WMMA_F32_16X16X4_F32:
//   A (16x4)  = activation chunk broadcast to all M rows (2 VGPRs / lane)
//   B (4x16)  = weight tile, lane n<-output row, kh selects K pair
//   C (16x16) = fp32 accumulator; row 0 lanes 0..15 hold the 16 dot products.
// Activations are DMA'd into LDS via GLOBAL_LOAD_ASYNC_TO_LDS_B128.
__global__ void gemv_wmma_kernel(const float* __restrict__ W, const float* __restrict__ x,
                                 const float* __restrict__ bias, float* __restrict__ y,
                                 int E, int Dm, int relu) {
    __shared__ float xs[FFNDIM];       // Dm <= 2048
    int tid = threadIdx.x;
    for (int i4 = tid; i4 < (Dm >> 2); i4 += blockDim.x)
        async_copy_b128(x + 4 * i4, xs + 4 * i4);
    async_wait0();
    __syncthreads();

    int wave = blockIdx.x * (blockDim.x >> 5) + (tid >> 5);
    int e0 = wave << 4;
    if (e0 >= E) return;               // wave-uniform

    int lane = tid & 31;
    int n = lane & 15;                 // output row within tile
    int kh = lane >> 4;                // K pair selector
    int e = e0 + n; if (e >= E) e = E - 1;     // clamp tail tile rows
    const float* __restrict__ Wr = W + (size_t)e * Dm;

    v8f acc = {0.f, 0.f, 0.f, 0.f, 0.f, 0.f, 0.f, 0.f};
    for (int k0 = 0; k0 < Dm; k0 += 4) {
        int kk = k0 + (kh << 1);
        if ((k0 & 127) == 0 && k0 + 256 < Dm)
            __builtin_prefetch(Wr + k0 + 256, 0, 1);   // global_prefetch_b8
        v2f a;  a[0] = xs[kk]; a[1] = xs[kk + 1];
        float2 wv = *(const float2*)(Wr + kk);
        v2f b2; b2[0] = wv.x; b2[1] = wv.y;
        acc = __builtin_amdgcn_wmma_f32_16x16x4_f32(
            /*neg_a=*/false, a, /*neg_b=*/false, b2,
            /*c_mod=*/(short)0, acc, /*reuse_a=*/false, /*reuse_b=*/false);
    }
    if (lane < 16 && (e0 + lane) < E) {
        float r = acc[0] + bias[e0 + lane];
        if (relu) r = fmaxf(r, 0.0f);
        y[e0 + lane] = r;
    }
}

// ---------------- attention: one block per head, 2048 positions ----------------
__global__ void attn_kernel(const float* __restrict__ q, const float* __restrict__ K,
                            const float* __restrict__ V, float* __restrict__ ctx) {
    __shared__ float sc[SKV];
    __shared__ float red[256];
    __shared__ float qs[HDIM];
    int h = blockIdx.x, tid = threadIdx.x;
    int off = h * HDIM;
    if (tid < HDIM / 4) async_copy_b128(q + off + 4 * tid, qs + 4 * tid);
    async_wait0();
    __syncthreads();

    const float scale = 0.17677669529663687f;  // 1/sqrt(32)
    float lmax = NEG_INF;
    for (int j = tid; j < SKV; j += 256) {
        const float* kr = K + (size_t)j * DMODEL + off;
        float d = 0.f;
#pragma unroll
        for (int c = 0; c < HDIM; c += 4) {
            float4 kv4 = *(const float4*)(kr + c);
            d += kv4.x * qs[c] + kv4.y * qs[c + 1] + kv4.z * qs[c + 2] + kv4.w * qs[c + 3];
        }
        d *= scale;
        sc[j] = d;
        lmax = fmaxf(lmax, d);
    }
    red[tid] = lmax; __syncthreads();
    for (int s = 128; s > 0; s >>= 1) {
        if (tid < s) red[tid] = fmaxf(red[tid], red[tid + s]);
        __syncthreads();
    }
    float m = red[0]; __syncthreads();

    float lsum = 0.f;
    for (int j = tid; j < SKV; j += 256) {
        float ex = __expf(sc[j] - m);
        sc[j] = ex; lsum += ex;
    }
    red[tid] = lsum; __syncthreads();
    for (int s = 128; s > 0; s >>= 1) {
        if (tid < s) red[tid] += red[tid + s];
        __syncthreads();
    }
    float inv = 1.0f / red[0]; __syncthreads();

    int d = tid & (HDIM - 1), g = tid >> 5;    // 8 position groups x 32 dims
    float acc = 0.f;
    for (int j = g; j < SKV; j += 8)
        acc += sc[j] * V[(size_t)j * DMODEL + off + d];
    red[tid] = acc; __syncthreads();
    if (tid < 128) red[tid] += red[tid + 128];
    __syncthreads();
    if (tid < 64) red[tid] += red[tid + 64];
    __syncthreads();
    if (tid < 32) red[tid] += red[tid + 32];
    __syncthreads();
    if (tid < HDIM) ctx[off + tid] = red[tid] * inv;
}

// ---------------- residual + layernorm, 512 threads ----------------
__global__ void resid_ln_kernel(const float* __restrict__ a, const float* __restrict__ b,
                                const float* __restrict__ g, const float* __restrict__ bb,
                                float* __restrict__ out) {
    __shared__ float r1[DMODEL];
    __shared__ float r2[DMODEL];
    int t = threadIdx.x;
    float v = a[t] + b[t];
    r1[t] = v; r2[t] = v * v; __syncthreads();
    for (int s = 256; s > 0; s >>= 1) {
        if (t < s) { r1[t] += r1[t + s]; r2[t] += r2[t + s]; }
        __syncthreads();
    }
    float mean = r1[0] * (1.0f / DMODEL);
    float var = r2[0] * (1.0f / DMODEL) - mean * mean;
    float rs = rsqrtf(var + 1e-5f);
    out[t] = (v - mean) * rs * g[t] + bb[t];
}

// ---------------- repetition penalty + top-k + softmax + noisy argmax ----------------
__global__ void sample_kernel(const float* __restrict__ logits, const int* __restrict__ y,
                              const float* __restrict__ noise,
                              float* __restrict__ ynew, float* __restrict__ samp) {
    __shared__ float lg[VOCAB];
    __shared__ float w[VOCAB];
    __shared__ float rv[256];
    __shared__ int   ri[256];
    __shared__ float pivot_s, maxv_s;
    int t = threadIdx.x;
    for (int i = t; i < VOCAB; i += 256) lg[i] = logits[i];
    __syncthreads();
    // repetition penalty on all history tokens (value based on original logits -> idempotent)
    for (int i = t; i < NTOK; i += 256) {
        int tok = y[i];
        float s0 = logits[tok];
        lg[tok] = (s0 < 0.f) ? s0 * 1.35f : s0 / 1.35f;
    }
    __syncthreads();
    for (int i = t; i < VOCAB; i += 256) w[i] = lg[i];
    __syncthreads();
    // 15 iterated argmax reductions -> pivot = 15th largest
    for (int it = 0; it < 15; ++it) {
        float bv = NEG_INF; int bi = VOCAB;
        for (int i = t; i < VOCAB; i += 256)
            if (w[i] > bv) { bv = w[i]; bi = i; }
        rv[t] = bv; ri[t] = bi; __syncthreads();
        for (int s = 128; s > 0; s >>= 1) {
            if (t < s) {
                if (rv[t + s] > rv[t] || (rv[t + s] == rv[t] && ri[t + s] < ri[t])) {
                    rv[t] = rv[t + s]; ri[t] = ri[t + s];
                }
            }
            __syncthreads();
        }
        if (t == 0) {
            pivot_s = rv[0];
            if (it == 0) maxv_s = rv[0];
            w[ri[0]] = NEG_INF;
        }
        __syncthreads();
    }
    float pivot = pivot_s, maxv = maxv_s;
    // mask below pivot, softmax
    float lsum = 0.f;
    for (int i = t; i < VOCAB; i += 256) {
        float v = (lg[i] < pivot) ? NEG_INF : lg[i];
        float e = (v == NEG_INF) ? 0.f : __expf(v - maxv);
        w[i] = e; lsum += e;
    }
    rv[t] = lsum; __syncthreads();
    for (int s = 128; s > 0; s >>= 1) {
        if (t < s) rv[t] += rv[t + s];
        __syncthreads();
    }
    float inv = 1.0f / rv[0]; __syncthreads();
    // argmax(probs / noise), first index wins ties
    float bv = NEG_INF; int bi = VOCAB;
    for (int i = t; i < VOCAB; i += 256) {
        float sc2 = (w[i] * inv) / noise[i];
        if (sc2 > bv) { bv = sc2; bi = i; }
    }
    rv[t] = bv; ri[t] = bi; __syncthreads();
    for (int s = 128; s > 0; s >>= 1) {
        if (t < s) {
            if (rv[t + s] > rv[t] || (rv[t + s] == rv[t] && ri[t + s] < ri[t])) {
                rv[t] = rv[t + s]; ri[t] = ri[t + s];
            }
        }
        __syncthreads();
    }
    if (t == 0) {
        ynew[NTOK] = (float)ri[0];
        samp[0] = (float)ri[0];
    }
}

// ---------------- host launch ----------------
extern "C" void kernel_launch(void* const* d_in, const int* in_sizes, int n_in,
                              void* d_out, int out_size, void* d_ws, size_t ws_size,
                              hipStream_t stream) {
    const int*   y         = (const int*)  d_in[0];
    const float* k_in      = (const float*)d_in[1];
    const float* v_in      = (const float*)d_in[2];
    const float* y_emb     = (const float*)d_in[3];
    // d_in[4] = x_example (unused)
    const float* alpha     = (const float*)d_in[5];
    const float* emb_table = (const float*)d_in[6];
    const float* in_proj_w = (const float*)d_in[7];
    const float* in_proj_b = (const float*)d_in[8];
    const float* out_w     = (const float*)d_in[9];
    const float* out_b     = (const float*)d_in[10];
    const float* ln1_w     = (const float*)d_in[11];
    const float* ln1_b     = (const float*)d_in[12];
    const float* ln2_w     = (const float*)d_in[13];
    const float* ln2_b     = (const float*)d_in[14];
    const float* ffn_w1    = (const float*)d_in[15];
    const float* ffn_b1    = (const float*)d_in[16];
    const float* ffn_w2    = (const float*)d_in[17];
    const float* ffn_b2    = (const float*)d_in[18];
    const float* pred_w    = (const float*)d_in[19];
    const float* pred_b    = (const float*)d_in[20];
    const float* noise     = (const float*)d_in[21];

    float* out = (float*)d_out;
    const size_t kv_elems  = (size_t)LAYERS * SKV * DMODEL;   // 25,165,824
    const size_t off_ynew  = 0;                               // 1537
    const size_t off_k     = 1537;
    const size_t off_v     = off_k + kv_elems;
    const size_t off_yemb  = off_v + kv_elems;                // 1536*512
    const size_t off_log   = off_yemb + (size_t)NTOK * DMODEL;
    const size_t off_samp  = off_log + VOCAB;
    (void)off_samp; (void)out_size; (void)ws_size; (void)n_in; (void)in_sizes;

    float* ws  = (float*)d_ws;
    float* x   = ws;            // 512
    float* qkv = ws + 512;      // 1536 (q = qkv[0:512])
    float* ctx = ws + 2048;     // 512
    float* ao  = ws + 2560;     // 512
    float* x1  = ws + 3072;     // 512
    float* hdn = ws + 3584;     // 2048
    float* ff  = ws + 5632;     // 512

    // prologue: token embed + PE, history copies, KV cache pad-copies
    prep_x_kernel<<<1, DMODEL, 0, stream>>>(y, emb_table, alpha, x,
                                            out + off_yemb + (size_t)TY * DMODEL);
    copy_i2f_kernel<<<6, 256, 0, stream>>>(y, out + off_ynew, NTOK);
    copy_f4_kernel<<<1024, 256, 0, stream>>>((const float4*)y_emb,
                                             (float4*)(out + off_yemb),
                                             (int)((size_t)TY * DMODEL / 4));
    copy_kv_kernel<<<2048, 256, 0, stream>>>((const float4*)k_in, (float4*)(out + off_k));
    copy_kv_kernel<<<2048, 256, 0, stream>>>((const float4*)v_in, (float4*)(out + off_v));

    for (int l = 0; l < LAYERS; ++l) {
        const size_t kvbase = (size_t)l * SKV * DMODEL;
        float* Kl = out + off_k + kvbase;
        float* Vl = out + off_v + kvbase;
        // qkv projection: 1536 outputs -> 96 waves -> 12 blocks
        gemv_wmma_kernel<<<12, 256, 0, stream>>>(in_proj_w + (size_t)l * 3 * DMODEL * DMODEL,
                                                 x, in_proj_b + (size_t)l * 3 * DMODEL,
                                                 qkv, 3 * DMODEL, DMODEL, 0);
        append_kv_kernel<<<2, DMODEL, 0, stream>>>(qkv,
                                                   Kl + (size_t)(SKV - 1) * DMODEL,
                                                   Vl + (size_t)(SKV - 1) * DMODEL);
        attn_kernel<<<HEADS, 256, 0, stream>>>(qkv, Kl, Vl, ctx);
        // out projection: 512 outputs -> 4 blocks
        gemv_wmma_kernel<<<4, 256, 0, stream>>>(out_w + (size_t)l * DMODEL * DMODEL,
                                                ctx, out_b + (size_t)l * DMODEL,
                                                ao, DMODEL, DMODEL, 0);
        resid_ln_kernel<<<1, DMODEL, 0, stream>>>(x, ao, ln1_w + (size_t)l * DMODEL,
                                                  ln1_b + (size_t)l * DMODEL, x1);
        // FFN1: 2048 outputs + relu -> 16 blocks
        gemv_wmma_kernel<<<16, 256, 0, stream>>>(ffn_w1 + (size_t)l * FFNDIM * DMODEL,
                                                 x1, ffn_b1 + (size_t)l * FFNDIM,
                                                 hdn, FFNDIM, DMODEL, 1);
        // FFN2: 512 outputs, K=2048 -> 4 blocks
        gemv_wmma_kernel<<<4, 256, 0, stream>>>(ffn_w2 + (size_t)l * DMODEL * FFNDIM,
                                                hdn, ffn_b2 + (size_t)l * DMODEL,
                                                ff, DMODEL, FFNDIM, 0);
        resid_ln_kernel<<<1, DMODEL, 0, stream>>>(x1, ff, ln2_w + (size_t)l * DMODEL,
                                                  ln2_b + (size_t)l * DMODEL, x);
    }

    // logits: 1025 outputs -> 65 waves -> 9 blocks
    gemv_wmma_kernel<<<9, 256, 0, stream>>>(pred_w, x, pred_b, out + off_log,
                                            VOCAB, DMODEL, 0);
    sample_kernel<<<1, 256, 0, stream>>>(out + off_log, y, noise,
                                         out + off_ynew, out + off_samp);
}